// OnehotEncoder_56633438765559
// MI455X (gfx1250) — compile-verified
//
#include <hip/hip_runtime.h>
#include <stdint.h>

// Problem constants (match the reference)
#define ENV_NUM 8192
#define EMB_DIM 128
#define BATCH   16384

typedef float v4f __attribute__((ext_vector_type(4)));

// ---------------------------------------------------------------------------
// CDNA5 async global->LDS copy (ASYNCcnt-tracked), per CDNA5 ISA
// VDST = LDS byte address (VGPR), VADDR = 64-bit global address (VGPR pair).
// ---------------------------------------------------------------------------
__device__ __forceinline__ void async_load_b32_to_lds(uint32_t lds_byte_off,
                                                      const void* gaddr) {
  asm volatile("global_load_async_to_lds_b32 %0, %1, off"
               :
               : "v"(lds_byte_off), "v"(gaddr)
               : "memory");
}

// Same, but non-temporal (source is streamed exactly once).
__device__ __forceinline__ void async_load_b32_to_lds_nt(uint32_t lds_byte_off,
                                                         const void* gaddr) {
  asm volatile("global_load_async_to_lds_b32 %0, %1, off th:TH_LOAD_NT"
               :
               : "v"(lds_byte_off), "v"(gaddr)
               : "memory");
}

__device__ __forceinline__ void wait_asynccnt0() {
  asm volatile("s_wait_asynccnt 0" ::: "memory");
}

// Low 32 bits of a generic pointer into LDS == LDS byte offset (ISA 10.2:
// LDS-aperture flat addresses use their low 32 bits as the LDS address).
__device__ __forceinline__ uint32_t lds_addr_of(const void* p) {
  return (uint32_t)(uintptr_t)p;
}

// ---------------------------------------------------------------------------
// Kernel 1: transpose W [EMB_DIM=128][ENV_NUM=8192] -> Wt [8192][128]
// 32x32 tiles through LDS; tile rows arrive via async global->LDS loads
// (coalesced 128B per wave, non-temporal: W is streamed once); transposed
// writes are coalesced 128B per wave. Padding to 33 floats/row makes the
// transposed LDS read bank-conflict free.
// ---------------------------------------------------------------------------
__global__ __launch_bounds__(256) void transpose_W_kernel(
    const float* __restrict__ W, float* __restrict__ Wt) {
  __shared__ float tile[32][33];

  const int tx = threadIdx.x;           // 0..31 (lane)
  const int ty = threadIdx.y;           // 0..7
  const int j0 = blockIdx.x * 32;       // env-dim tile origin
  const int e0 = blockIdx.y * 32;       // emb-dim tile origin

#pragma unroll
  for (int k = 0; k < 4; ++k) {
    const int r = ty + k * 8;           // row within tile (emb dim)
    async_load_b32_to_lds_nt(lds_addr_of(&tile[r][tx]),
                             W + (size_t)(e0 + r) * ENV_NUM + (j0 + tx));
  }
  wait_asynccnt0();                     // our writes landed in LDS
  __syncthreads();                      // everyone's writes visible

#pragma unroll
  for (int k = 0; k < 4; ++k) {
    const int r = ty + k * 8;           // row within Wt tile (env dim)
    Wt[(size_t)(j0 + r) * EMB_DIM + (e0 + tx)] = tile[tx][r];
  }
}

// ---------------------------------------------------------------------------
// Kernel 2: out[b][:] = Wt[idx[b]][:] + bias   (fully coalesced float4 path)
// One wave32 per output row: 32 lanes x float4 = 128 floats.
// Bias staged into LDS via the async pipe; output stores are non-temporal
// (written once, never re-read on device).
// ---------------------------------------------------------------------------
__global__ __launch_bounds__(256) void gather_bias_kernel(
    const int* __restrict__ idx, const float* __restrict__ Wt,
    const float* __restrict__ bias, float* __restrict__ out) {
  __shared__ float bias_s[EMB_DIM];

  const int t = threadIdx.x;
  if (t < EMB_DIM) {
    async_load_b32_to_lds(lds_addr_of(&bias_s[t]), bias + t);
  }
  wait_asynccnt0();
  __syncthreads();

  const int row  = blockIdx.x * 8 + (t >> 5);  // 8 waves -> 8 rows per block
  const int lane = t & 31;
  const int j    = idx[row];

  const v4f w  = *(const v4f*)(Wt + (size_t)j * EMB_DIM + lane * 4);
  const v4f bb = *(const v4f*)(&bias_s[lane * 4]);
  const v4f o  = w + bb;
  __builtin_nontemporal_store(o, (v4f*)(out + (size_t)row * EMB_DIM + lane * 4));
}

// ---------------------------------------------------------------------------
// Fallback (workspace too small): direct strided column gather from W.
// ---------------------------------------------------------------------------
__global__ __launch_bounds__(256) void gather_direct_kernel(
    const int* __restrict__ idx, const float* __restrict__ W,
    const float* __restrict__ bias, float* __restrict__ out) {
  const int t    = threadIdx.x;
  const int row  = blockIdx.x * 8 + (t >> 5);
  const int lane = t & 31;
  const int j    = idx[row];
  const int e    = lane * 4;

  v4f o;
  o.x = W[(size_t)(e + 0) * ENV_NUM + j] + bias[e + 0];
  o.y = W[(size_t)(e + 1) * ENV_NUM + j] + bias[e + 1];
  o.z = W[(size_t)(e + 2) * ENV_NUM + j] + bias[e + 2];
  o.w = W[(size_t)(e + 3) * ENV_NUM + j] + bias[e + 3];
  __builtin_nontemporal_store(o, (v4f*)(out + (size_t)row * EMB_DIM + e));
}

// ---------------------------------------------------------------------------
extern "C" void kernel_launch(void* const* d_in, const int* in_sizes, int n_in,
                              void* d_out, int out_size, void* d_ws,
                              size_t ws_size, hipStream_t stream) {
  (void)in_sizes; (void)n_in; (void)out_size;

  const int*   idx  = (const int*)d_in[0];     // noise_idx  [16384] int32
  const float* W    = (const float*)d_in[1];   // W          [128, 8192] f32
  const float* bias = (const float*)d_in[2];   // b          [128] f32
  float*       out  = (float*)d_out;           // [16384, 1, 128] f32

  const size_t wt_bytes = (size_t)ENV_NUM * EMB_DIM * sizeof(float);  // 4 MB

  if (ws_size >= wt_bytes) {
    float* Wt = (float*)d_ws;
    transpose_W_kernel<<<dim3(ENV_NUM / 32, EMB_DIM / 32), dim3(32, 8), 0,
                         stream>>>(W, Wt);
    gather_bias_kernel<<<BATCH / 8, 256, 0, stream>>>(idx, Wt, bias, out);
  } else {
    gather_direct_kernel<<<BATCH / 8, 256, 0, stream>>>(idx, W, bias, out);
  }
}